// LiftSplatShoot_13434657702314
// MI455X (gfx1250) — compile-verified
//
#include <hip/hip_runtime.h>
#include <hip/hip_bf16.h>
#include <stdint.h>

// ---------------- problem constants (from reference) ----------------
#define BATCH   8
#define NCAM    6
#define ND      41
#define FH      14
#define FW      30
#define CC      64
#define NX0     200
#define NX1     200
#define NZV     1
#define PLANE   (NX0 * NX1)            // 40000
#define OUT_PER_B (CC * PLANE)         // 2,560,000
#define NPOINTS (BATCH * NCAM * ND * FH * FW)  // 826,560

#define SCATTER_BLOCKS  4096
#define SCATTER_THREADS 256
#define TOTAL_WAVES     ((SCATTER_BLOCKS * SCATTER_THREADS) / 32)   // 32768
#define CHUNK           ((NPOINTS + TOTAL_WAVES - 1) / TOTAL_WAVES) // 26

typedef float v2f __attribute__((ext_vector_type(2)));
typedef float v4f __attribute__((ext_vector_type(4)));

// ---------------- codegen probe (NEVER LAUNCHED) --------------------------------
// First function in the file so the disasm snippet shows exactly how the three
// critical ops lower on gfx1250: NT load hint, global_prefetch_b8, and native
// global_atomic_add_f32 (vs a cmpswap loop).
__global__ void lss_probe_atomic(float* __restrict__ out, const float* __restrict__ in) {
    int i = threadIdx.x;
    float v = __builtin_nontemporal_load(in + i);
    __builtin_prefetch(in + i + 64, 0, 1);
    atomicAdd(out + (size_t)i * PLANE, v);
}

// ---------------- main scatter-add ----------------------------------------------
// One wave32 per point. Lane L handles channels 2L, 2L+1 (float2 coalesced load:
// 256B per wave per point). Each wave owns a CONTIGUOUS chunk of ~26 points:
//  - (b,n,d,h,w) is decomposed once and carried incrementally (uniform branches)
//  - the 12-float camera matrix lives in VGPRs, reloaded only when the camera
//    changes (at most once per chunk)
// Atomic f32 adds land in the 192MB L2 (output = 82MB, L2-resident).
__global__ __launch_bounds__(SCATTER_THREADS) void lss_scatter_kernel(
        const float* __restrict__ x,
        const float* __restrict__ comb,
        float* __restrict__ out) {
    const int lane = threadIdx.x & 31;
    const int wave = (blockIdx.x * blockDim.x + threadIdx.x) >> 5;

    int p0 = wave * CHUNK;
    int p1 = p0 + CHUNK;
    if (p1 > NPOINTS) p1 = NPOINTS;
    if (p0 >= p1) return;

    // one-time decomposition of p0 -> (b, n, d, h, w)
    int t = p0;
    int w = t % FW;  t /= FW;
    int h = t % FH;  t /= FH;
    int d = t % ND;  t /= ND;
    int n = t % NCAM;
    int b = t / NCAM;

    // camera matrix cached in registers
    float m0, m1, m2, m3, m4, m5, m6, m7, m8, tx, ty, tz;
    {
        const float* cm = comb + (size_t)(b * NCAM + n) * 12;
        m0 = cm[0]; m1 = cm[1]; m2 = cm[2];
        m3 = cm[3]; m4 = cm[4]; m5 = cm[5];
        m6 = cm[6]; m7 = cm[7]; m8 = cm[8];
        tx = cm[9]; ty = cm[10]; tz = cm[11];
    }

    const size_t laneChanOff = (size_t)(2 * lane) * PLANE;   // loop-invariant

    for (int p = p0; p < p1; ++p) {
        // Prefetch next point's 256B x block (read-once stream).
        if (p + 1 < p1) {
            __builtin_prefetch(x + (size_t)(p + 1) * CC + 2 * lane, 0, 1);
        }

        // frustum pixel coords: xs=linspace(0,479,30), ys=linspace(0,223,14)
        float xs = (float)w * (479.0f / 29.0f);
        float ys = (float)h * (223.0f / 13.0f);
        float dv = 4.0f + (float)d;

        float px = xs * dv, py = ys * dv, pz = dv;

        float gxf = m0 * px + m1 * py + m2 * pz + tx;
        float gyf = m3 * px + m4 * py + m5 * pz + ty;
        float gzf = m6 * px + m7 * py + m8 * pz + tz;

        // voxelize: floor((g - lower)/dx); lower = (-50,-50,-10), dx=(0.5,0.5,20)
        float fgx = floorf((gxf + 50.0f) / 0.5f);
        float fgy = floorf((gyf + 50.0f) / 0.5f);
        float fgz = floorf((gzf + 10.0f) / 20.0f);

        bool valid = (fgx >= 0.0f) & (fgx < (float)NX0) &
                     (fgy >= 0.0f) & (fgy < (float)NX1) &
                     (fgz >= 0.0f) & (fgz < (float)NZV);

        if (valid) {  // wave-uniform branch
            int gx = (int)fgx;
            int gy = (int)fgy;
            // gz == 0 when valid (NZV == 1)
            float* dst = out + (size_t)b * OUT_PER_B + (size_t)gx * NX1 + (size_t)gy
                             + laneChanOff;

            // non-temporal: keep the 211MB x stream out of L2 so the
            // 82MB output stays resident for the atomics
            const v2f* xp = (const v2f*)(x + (size_t)p * CC) + lane;
            v2f v = __builtin_nontemporal_load(xp);

            atomicAdd(dst,         v.x);
            atomicAdd(dst + PLANE, v.y);   // +160000B: fits the 24-bit imm offset
        }

        // incremental (b,n,d,h,w) update — all branches wave-uniform
        if (++w == FW) {
            w = 0;
            if (++h == FH) {
                h = 0;
                if (++d == ND) {
                    d = 0;
                    if (++n == NCAM) { n = 0; ++b; }
                    const float* cm = comb + (size_t)(b * NCAM + n) * 12;
                    m0 = cm[0]; m1 = cm[1]; m2 = cm[2];
                    m3 = cm[3]; m4 = cm[4]; m5 = cm[5];
                    m6 = cm[6]; m7 = cm[7]; m8 = cm[8];
                    tx = cm[9]; ty = cm[10]; tz = cm[11];
                }
            }
        }
    }
}

// ---------------- zero-fill output (d_out is poisoned by harness) ----------------
__global__ __launch_bounds__(256) void lss_zero_kernel(float* __restrict__ out, int n) {
    int i = blockIdx.x * blockDim.x + threadIdx.x;
    int i4 = i * 4;
    if (i4 + 3 < n) {
        *(v4f*)(out + i4) = (v4f){0.f, 0.f, 0.f, 0.f};
    }
    if (i == 0) {  // tail (n % 4 elements; n is actually a multiple of 4)
        for (int k = n & ~3; k < n; ++k) out[k] = 0.f;
    }
}

// ---------------- setup: comb[b][n] = R * inv(K), trans -------------------------
// K: (B,N,3,3) row-major, E: (B,N,4,4) row-major
// ws layout: 12 floats per (b,n): comb row-major [0..8], trans [9..11]
__global__ __launch_bounds__(64) void lss_setup_kernel(const float* __restrict__ K,
                                                       const float* __restrict__ E,
                                                       float* __restrict__ comb) {
    int i = threadIdx.x;
    if (i >= BATCH * NCAM) return;
    const float* k = K + (size_t)i * 9;
    const float* e = E + (size_t)i * 16;

    float a = k[0], b = k[1], c = k[2];
    float d = k[3], f = k[4], g = k[5];
    float h = k[6], p = k[7], q = k[8];

    float A =  (f * q - g * p);
    float Bc = -(d * q - g * h);
    float Cc =  (d * p - f * h);
    float det = a * A + b * Bc + c * Cc;
    float r = 1.0f / det;

    float inv[9];
    inv[0] = A * r;   inv[1] = -(b * q - c * p) * r;  inv[2] =  (b * g - c * f) * r;
    inv[3] = Bc * r;  inv[4] =  (a * q - c * h) * r;  inv[5] = -(a * g - c * d) * r;
    inv[6] = Cc * r;  inv[7] = -(a * p - b * h) * r;  inv[8] =  (a * f - b * d) * r;

    float o[12];
    #pragma unroll
    for (int rr = 0; rr < 3; ++rr) {
        #pragma unroll
        for (int cc = 0; cc < 3; ++cc) {
            float s = 0.f;
            #pragma unroll
            for (int t = 0; t < 3; ++t) s += e[rr * 4 + t] * inv[t * 3 + cc];
            o[rr * 3 + cc] = s;
        }
    }
    o[9]  = e[0 * 4 + 3];
    o[10] = e[1 * 4 + 3];
    o[11] = e[2 * 4 + 3];

    #pragma unroll
    for (int t = 0; t < 12; ++t) comb[i * 12 + t] = o[t];
}

// ---------------- launcher ----------------
extern "C" void kernel_launch(void* const* d_in, const int* in_sizes, int n_in,
                              void* d_out, int out_size, void* d_ws, size_t ws_size,
                              hipStream_t stream) {
    const float* x = (const float*)d_in[0];   // (B,N,D,fH,fW,C) f32
    const float* K = (const float*)d_in[1];   // (B,N,3,3) f32
    const float* E = (const float*)d_in[2];   // (B,N,4,4) f32
    float* out = (float*)d_out;               // (B, C, 200, 200) f32
    float* comb = (float*)d_ws;               // 48 * 12 floats

    // 1) zero output (poisoned by harness)
    int n4 = (out_size + 3) / 4;
    int zblocks = (n4 + 255) / 256;
    lss_zero_kernel<<<zblocks, 256, 0, stream>>>(out, out_size);

    // 2) per-camera comb = R * inv(K), trans
    lss_setup_kernel<<<1, 64, 0, stream>>>(K, E, comb);

    // 3) scatter-add: each wave owns a contiguous ~26-point chunk
    lss_scatter_kernel<<<SCATTER_BLOCKS, SCATTER_THREADS, 0, stream>>>(x, comb, out);
}